// GraphAttentionPosEnc_7043746365720
// MI455X (gfx1250) — compile-verified
//
#include <hip/hip_runtime.h>
#include <math.h>

typedef __attribute__((ext_vector_type(2))) float v2f;
typedef __attribute__((ext_vector_type(8))) float v8f;

#define N_HEADS   4
#define HEAD_DIM  16
#define OUT_DIM   64
#define IN_DIM    128
#define NEG_SLOPE 0.2f

// ---------------------------------------------------------------------------
// Ordered-uint mapping so unsigned atomicMax implements float max.
// ---------------------------------------------------------------------------
__device__ __forceinline__ unsigned f2ord(float f) {
    unsigned u = __float_as_uint(f);
    return (u & 0x80000000u) ? ~u : (u | 0x80000000u);
}
__device__ __forceinline__ float ord2f(unsigned u) {
    return (u & 0x80000000u) ? __uint_as_float(u & 0x7FFFFFFFu)
                             : __uint_as_float(~u);
}
__device__ __forceinline__ float leaky(float x) {
    return x >= 0.0f ? x : NEG_SLOPE * x;
}

// ---------------------------------------------------------------------------
// Kernel 1: h = [x|state] @ W^T via V_WMMA_F32_16X16X4_F32.
// One wave computes a 16-row strip across all 4 column tiles (heads).
// W (64x128, 32KB) staged in LDS with pad-130 stride to avoid bank conflicts.
// ---------------------------------------------------------------------------
#define W_LDS_STRIDE 130

__global__ __launch_bounds__(256) void gemm_wmma_kernel(
    const float* __restrict__ x, const float* __restrict__ state,
    const float* __restrict__ W, float* __restrict__ h, int n_nodes)
{
    __shared__ float Wlds[OUT_DIM * W_LDS_STRIDE];  // 33,280 B

    // Cooperative load of W[n][k] (row-major 64x128) into padded LDS.
    for (int i = threadIdx.x; i < OUT_DIM * IN_DIM; i += blockDim.x) {
        int n = i >> 7, k = i & 127;
        Wlds[n * W_LDS_STRIDE + k] = W[i];
    }
    __syncthreads();

    const int wave = threadIdx.x >> 5;
    const int lane = threadIdx.x & 31;
    const int m0   = (blockIdx.x * 8 + wave) * 16;
    if (m0 >= n_nodes) return;

    const int col = lane & 15;       // N within tile / M row for A frag
    const int hi  = lane >> 4;       // half-wave selector
    const int khi = hi * 2;          // K offset for this half (A & B frags)

    int rowa = m0 + col;
    if (rowa >= n_nodes) rowa = n_nodes - 1;        // safe clamp (stores guarded)
    const float* __restrict__ xrow = x     + (size_t)rowa * 64;
    const float* __restrict__ srow = state + (size_t)rowa * 64;

    v8f acc[4];
#pragma unroll
    for (int j = 0; j < 4; ++j) acc[j] = (v8f){0,0,0,0,0,0,0,0};

#pragma unroll 4
    for (int k0 = 0; k0 < IN_DIM; k0 += 4) {
        const int kk = k0 + khi;                    // this lane's K pair
        const float* ap = (kk < 64) ? (xrow + kk) : (srow + (kk - 64));
        v2f a; a.x = ap[0]; a.y = ap[1];
#pragma unroll
        for (int j = 0; j < 4; ++j) {
            const float* bp = &Wlds[(j * 16 + col) * W_LDS_STRIDE + kk];
            v2f b; b.x = bp[0]; b.y = bp[1];
            acc[j] = __builtin_amdgcn_wmma_f32_16x16x4_f32(
                false, a, false, b, (short)0, acc[j], false, false);
        }
    }

    // D layout: VGPR r holds M = r + 8*hi, N = col (within tile j).
    // Wave-uniform guard: full tiles take the unguarded coalesced path.
    float* __restrict__ hout = h + (size_t)(m0 + 8 * hi) * OUT_DIM + col;
    if (m0 + 16 <= n_nodes) {
#pragma unroll
        for (int j = 0; j < 4; ++j) {
#pragma unroll
            for (int r = 0; r < 8; ++r)
                hout[(size_t)r * OUT_DIM + j * 16] = acc[j][r];
        }
    } else {
#pragma unroll
        for (int j = 0; j < 4; ++j) {
#pragma unroll
            for (int r = 0; r < 8; ++r) {
                int m = m0 + r + 8 * hi;
                if (m < n_nodes)
                    h[(size_t)m * OUT_DIM + j * 16 + col] = acc[j][r];
            }
        }
    }
}

// ---------------------------------------------------------------------------
// Kernel 2: per-(node,head) attention scores. thread t = node*4 + head.
// ---------------------------------------------------------------------------
__global__ __launch_bounds__(256) void score_kernel(
    const float* __restrict__ h,
    const float* __restrict__ attn_src, const float* __restrict__ attn_dst,
    float* __restrict__ s_src, float* __restrict__ s_dst, int n_nodes)
{
    int t = blockIdx.x * blockDim.x + threadIdx.x;
    if (t >= n_nodes * N_HEADS) return;
    int head = t & 3;
    int node = t >> 2;
    const float4* hp = (const float4*)(h + (size_t)node * OUT_DIM + head * HEAD_DIM);
    const float4* as = (const float4*)(attn_src + head * HEAD_DIM);
    const float4* ad = (const float4*)(attn_dst + head * HEAD_DIM);
    float ss = 0.0f, sd = 0.0f;
#pragma unroll
    for (int q = 0; q < 4; ++q) {
        float4 hv = hp[q], av = as[q], dv = ad[q];
        ss += hv.x * av.x + hv.y * av.y + hv.z * av.z + hv.w * av.w;
        sd += hv.x * dv.x + hv.y * dv.y + hv.z * dv.z + hv.w * dv.w;
    }
    s_src[t] = ss;
    s_dst[t] = sd;
}

// ---------------------------------------------------------------------------
// Kernel 3: segment-max of leaky(logits) per dst node via ordered-uint atomicMax.
// ---------------------------------------------------------------------------
__global__ __launch_bounds__(256) void edge_max_kernel(
    const int* __restrict__ srcs, const int* __restrict__ dsts,
    const float* __restrict__ s_src, const float* __restrict__ s_dst,
    unsigned* __restrict__ maxbits, int n_edges)
{
    int e = blockIdx.x * blockDim.x + threadIdx.x;
    if (e >= n_edges) return;
    int s = srcs[e], d = dsts[e];
    float4 ss = *(const float4*)(s_src + (size_t)s * 4);
    float4 sd = *(const float4*)(s_dst + (size_t)d * 4);
    unsigned* mb = maxbits + (size_t)d * 4;
    atomicMax(mb + 0, f2ord(leaky(ss.x + sd.x)));
    atomicMax(mb + 1, f2ord(leaky(ss.y + sd.y)));
    atomicMax(mb + 2, f2ord(leaky(ss.z + sd.z)));
    atomicMax(mb + 3, f2ord(leaky(ss.w + sd.w)));
}

// ---------------------------------------------------------------------------
// Kernel 4: softmax denominator per dst node (exp shifted by segment max).
// ---------------------------------------------------------------------------
__global__ __launch_bounds__(256) void edge_sum_kernel(
    const int* __restrict__ srcs, const int* __restrict__ dsts,
    const float* __restrict__ s_src, const float* __restrict__ s_dst,
    const unsigned* __restrict__ maxbits, float* __restrict__ denom, int n_edges)
{
    int e = blockIdx.x * blockDim.x + threadIdx.x;
    if (e >= n_edges) return;
    int s = srcs[e], d = dsts[e];
    float4 ss = *(const float4*)(s_src + (size_t)s * 4);
    float4 sd = *(const float4*)(s_dst + (size_t)d * 4);
    uint4  mb = *(const uint4*)(maxbits + (size_t)d * 4);
    float* dn = denom + (size_t)d * 4;
    atomicAdd(dn + 0, expf(leaky(ss.x + sd.x) - ord2f(mb.x)));
    atomicAdd(dn + 1, expf(leaky(ss.y + sd.y) - ord2f(mb.y)));
    atomicAdd(dn + 2, expf(leaky(ss.z + sd.z) - ord2f(mb.z)));
    atomicAdd(dn + 3, expf(leaky(ss.w + sd.w) - ord2f(mb.w)));
}

// ---------------------------------------------------------------------------
// Kernel 5: weighted aggregation. 64 consecutive lanes serve one edge
// (lane -> head*16+d) so the h[src] gather and the atomic scatter into out
// are fully coalesced 256B transactions (h and out are L2-resident).
// ---------------------------------------------------------------------------
__global__ __launch_bounds__(256) void aggregate_kernel(
    const int* __restrict__ srcs, const int* __restrict__ dsts,
    const float* __restrict__ h,
    const float* __restrict__ s_src, const float* __restrict__ s_dst,
    const unsigned* __restrict__ maxbits, const float* __restrict__ denom,
    float* __restrict__ out, int n_edges)
{
    int t = blockIdx.x * blockDim.x + threadIdx.x;   // < n_edges*64 < 2^31
    int e = t >> 6;
    if (e >= n_edges) return;
    int c    = t & 63;
    int head = c >> 4;
    int s = srcs[e], d = dsts[e];
    float l  = leaky(s_src[(size_t)s * 4 + head] + s_dst[(size_t)d * 4 + head]);
    float mx = ord2f(maxbits[(size_t)d * 4 + head]);
    float alpha = expf(l - mx) / (denom[(size_t)d * 4 + head] + 1e-12f);
    float val = h[(size_t)s * OUT_DIM + c] * alpha;
    atomicAdd(&out[(size_t)d * OUT_DIM + c], val);
}

// ---------------------------------------------------------------------------
// Kernel 6: residual + ELU.
// ---------------------------------------------------------------------------
__global__ __launch_bounds__(256) void finalize_kernel(
    float* __restrict__ out, const float* __restrict__ h, int n)
{
    int i = blockIdx.x * blockDim.x + threadIdx.x;
    if (i >= n) return;
    float v = out[i] + h[i];
    out[i] = v > 0.0f ? v : expm1f(v);
}

// ---------------------------------------------------------------------------
extern "C" void kernel_launch(void* const* d_in, const int* in_sizes, int n_in,
                              void* d_out, int out_size, void* d_ws, size_t ws_size,
                              hipStream_t stream) {
    const float* x        = (const float*)d_in[0];
    const float* state    = (const float*)d_in[1];
    const int*   eidx     = (const int*)d_in[2];
    // d_in[3] = edge_weight, ignored by design
    const float* W        = (const float*)d_in[4];
    const float* attn_src = (const float*)d_in[5];
    const float* attn_dst = (const float*)d_in[6];

    const int n_nodes = in_sizes[0] / 64;
    const int n_edges = in_sizes[2] / 2;
    const int* srcs = eidx;
    const int* dsts = eidx + n_edges;

    // Workspace layout (all 16B aligned for the sizes in play).
    float*    h       = (float*)d_ws;                              // n_nodes*64
    float*    s_src   = h + (size_t)n_nodes * OUT_DIM;             // n_nodes*4
    float*    s_dst   = s_src + (size_t)n_nodes * N_HEADS;         // n_nodes*4
    unsigned* maxbits = (unsigned*)(s_dst + (size_t)n_nodes * N_HEADS); // n_nodes*4
    float*    denom   = (float*)(maxbits + (size_t)n_nodes * N_HEADS);  // n_nodes*4

    // Zero the accumulators (graph-capture safe).
    hipMemsetAsync(d_out, 0, (size_t)out_size * sizeof(float), stream);
    hipMemsetAsync(maxbits, 0, (size_t)n_nodes * N_HEADS * sizeof(unsigned), stream);
    hipMemsetAsync(denom, 0, (size_t)n_nodes * N_HEADS * sizeof(float), stream);

    const int gemm_blocks = (n_nodes + 127) / 128;   // 8 waves x 16 rows per block
    gemm_wmma_kernel<<<gemm_blocks, 256, 0, stream>>>(x, state, W, h, n_nodes);

    score_kernel<<<(n_nodes * N_HEADS + 255) / 256, 256, 0, stream>>>(
        h, attn_src, attn_dst, s_src, s_dst, n_nodes);

    edge_max_kernel<<<(n_edges + 255) / 256, 256, 0, stream>>>(
        srcs, dsts, s_src, s_dst, maxbits, n_edges);

    edge_sum_kernel<<<(n_edges + 255) / 256, 256, 0, stream>>>(
        srcs, dsts, s_src, s_dst, maxbits, denom, n_edges);

    const long long total = (long long)n_edges * OUT_DIM;
    aggregate_kernel<<<(int)((total + 255) / 256), 256, 0, stream>>>(
        srcs, dsts, h, s_src, s_dst, maxbits, denom, (float*)d_out, n_edges);

    finalize_kernel<<<(n_nodes * OUT_DIM + 255) / 256, 256, 0, stream>>>(
        (float*)d_out, h, n_nodes * OUT_DIM);
}